// QuantumNet_78082505441471
// MI455X (gfx1250) — compile-verified
//
#include <hip/hip_runtime.h>

namespace {

typedef unsigned u32;
typedef u32 u32x4 __attribute__((ext_vector_type(4)));
typedef u32 u32x8 __attribute__((ext_vector_type(8)));

#if __has_builtin(__builtin_amdgcn_s_wait_tensorcnt)
#define WAIT_TENSORCNT(n) __builtin_amdgcn_s_wait_tensorcnt(n)
#else
#define WAIT_TENSORCNT(n) asm volatile("s_wait_tensorcnt %0" ::"i"(n))
#endif

struct Cx { float re, im; };

__device__ __forceinline__ Cx cmul(Cx a, Cx b) {
    return Cx{ a.re * b.re - a.im * b.im, a.re * b.im + a.im * b.re };
}
__device__ __forceinline__ Cx cconj(Cx a) { return Cx{ a.re, -a.im }; }

// RX mixing: new0 = c*t0 + (-i s)*t1 ; new1 = (-i s)*t0 + c*t1
__device__ __forceinline__ void rx_mix(Cx& t0, Cx& t1, float c, float s) {
    Cx n0{ c * t0.re + s * t1.im, c * t0.im - s * t1.re };
    Cx n1{ c * t1.re + s * t0.im, c * t1.im - s * t0.re };
    t0 = n0; t1 = n1;
}

// One wave, runs once: build fixed 4x4 real symmetric T with out = u^T T u,
// u = (c0c1, c0s1, s0c1, s0s1).  T = Re( conj(d_j) d_k (U^H P0 U)_jk ),
// d = (1,-i,-i,-1), P0 = projector onto q0=0.
__global__ void qnet_build_T(const float* __restrict__ qw, float* __restrict__ T) {
    if (threadIdx.x != 0 || blockIdx.x != 0) return;
    float c[2][2], s[2][2];
    for (int l = 0; l < 2; ++l)
        for (int q = 0; q < 2; ++q) {
            float w = 0.5f * qw[l * 2 + q];
            c[l][q] = cosf(w);
            s[l][q] = sinf(w);
        }
    Cx U[4][4];
    for (int j = 0; j < 4; ++j) {
        Cx a[4] = { {0.f,0.f},{0.f,0.f},{0.f,0.f},{0.f,0.f} };
        a[j].re = 1.0f;
        for (int l = 0; l < 2; ++l) {
            rx_mix(a[0], a[2], c[l][0], s[l][0]);   // RX q0, col 0
            rx_mix(a[1], a[3], c[l][0], s[l][0]);   // RX q0, col 1
            rx_mix(a[0], a[1], c[l][1], s[l][1]);   // RX q1, row 0
            rx_mix(a[2], a[3], c[l][1], s[l][1]);   // RX q1, row 1
            Cx t = a[2]; a[2] = a[3]; a[3] = t;     // CNOT(0->1)
        }
        for (int i = 0; i < 4; ++i) U[i][j] = a[i];
    }
    const Cx d[4] = { {1.f,0.f}, {0.f,-1.f}, {0.f,-1.f}, {-1.f,0.f} };
    for (int j = 0; j < 4; ++j)
        for (int k = 0; k < 4; ++k) {
            Cx m{0.f, 0.f};
            for (int i = 0; i < 2; ++i) {           // P0 rows |00>, |01>
                Cx t = cmul(cconj(U[i][j]), U[i][k]);
                m.re += t.re; m.im += t.im;
            }
            Cx b = cmul(cconj(d[j]), cmul(d[k], m));
            T[j * 4 + k] = b.re;
        }
}

__device__ __forceinline__ float eval_sample(float x0, float x1,
                                             const float4& T0, const float4& T1,
                                             const float4& T2, const float4& T3) {
    float c0 = __cosf(0.5f * x0), s0 = __sinf(0.5f * x0);
    float c1 = __cosf(0.5f * x1), s1 = __sinf(0.5f * x1);
    float u0 = c0 * c1, u1 = c0 * s1, u2 = s0 * c1, u3 = s0 * s1;
    float t0 = T0.x * u0 + T0.y * u1 + T0.z * u2 + T0.w * u3;
    float t1 = T1.x * u0 + T1.y * u1 + T1.z * u2 + T1.w * u3;
    float t2 = T2.x * u0 + T2.y * u1 + T2.z * u2 + T2.w * u3;
    float t3 = T3.x * u0 + T3.y * u1 + T3.z * u2 + T3.w * u3;
    return u0 * t0 + u1 * t1 + u2 * t2 + u3 * t3;   // = P(q0=0)
}

// ---------------- TDM streaming kernel ----------------
// 256 threads = 8 waves. Each wave DMA-stages 1 KB tiles of x (256 floats =
// 128 samples) into its private LDS double buffer via tensor_load_to_lds,
// double-buffered on TENSORcnt; 4 tiles per wave -> 4096 samples per block.
__global__ __launch_bounds__(256) void qnet_main_tdm(const float* __restrict__ x,
                                                     const float* __restrict__ T,
                                                     float* __restrict__ out) {
    __shared__ float4 stage[2][8][64];               // 2 bufs x 8 waves x 1KB = 16KB

    const u32 lane = threadIdx.x & 31u;
    const u32 w = (u32)__builtin_amdgcn_readfirstlane((int)(threadIdx.x >> 5));

    const float4* Tv = (const float4*)T;
    const float4 T0 = Tv[0], T1 = Tv[1], T2 = Tv[2], T3 = Tv[3];

    // wave's first sample: block*4096 + wave*512 ; iter t adds t*128 samples
    const unsigned long long waveSample =
        (unsigned long long)blockIdx.x * 4096ull + (unsigned long long)w * 512ull;
    const unsigned long long gbase =
        (unsigned long long)(size_t)x + waveSample * 8ull;    // 2 floats/sample
    const u32 ldsBase = (u32)(unsigned long long)(size_t)&stage[0][0][0];

    auto issue = [&](u32 t, u32 b) {
        unsigned long long ga = gbase + (unsigned long long)t * 1024ull;
        u32x4 g0;
        g0.x = 1u;                                    // count=1, user descriptor
        g0.y = ldsBase + (b * 8u + w) * 1024u;        // lds_addr (bytes)
        g0.z = (u32)ga;                               // global_addr[31:0]
        g0.w = ((u32)(ga >> 32) & 0x1FFFFFFu) | 0x80000000u;  // addr[56:32] | type=2
        u32x8 g1;
        g1[0] = 2u << 16;                             // data_size = 4B, no multicast
        g1[1] = 256u << 16;                           // tensor_dim0 = 256 elements
        g1[2] = 1u << 16;                             // tensor_dim1 = 1
        g1[3] = 256u << 16;                           // tile_dim0 = 256 (1 KB row)
        g1[4] = 0u;                                   // tile_dim1/2 unused
        g1[5] = 256u;                                 // tensor_dim0_stride = 256
        g1[6] = 0u;
        g1[7] = 0u;
        asm volatile("tensor_load_to_lds %0, %1" :: "s"(g0), "s"(g1) : "memory");
    };

    issue(0u, 0u);
    #pragma unroll
    for (u32 t = 0; t < 4u; ++t) {
        const u32 cur = t & 1u;
        if (t < 3u) {
            issue(t + 1u, cur ^ 1u);                  // prefetch next tile
            WAIT_TENSORCNT(1);                        // tile t landed in LDS
        } else {
            WAIT_TENSORCNT(0);
        }
        asm volatile("" ::: "memory");
        const float4 xa = stage[cur][w][lane * 2u];       // samples .. +0,+1
        const float4 xb = stage[cur][w][lane * 2u + 1u];  // samples .. +2,+3
        float4 r;
        r.x = eval_sample(xa.x, xa.y, T0, T1, T2, T3);
        r.y = eval_sample(xa.z, xa.w, T0, T1, T2, T3);
        r.z = eval_sample(xb.x, xb.y, T0, T1, T2, T3);
        r.w = eval_sample(xb.z, xb.w, T0, T1, T2, T3);
        ((float4*)out)[(waveSample >> 2) + t * 32u + lane] = r;
    }
}

// ---------------- Fallback: direct b128 streaming (any B) ----------------
__global__ __launch_bounds__(256) void qnet_main(const float* __restrict__ x,
                                                 const float* __restrict__ T,
                                                 float* __restrict__ out, int B) {
    const int tid = blockIdx.x * 256 + threadIdx.x;
    const float4* Tv = (const float4*)T;
    const float4 T0 = Tv[0], T1 = Tv[1], T2 = Tv[2], T3 = Tv[3];

    const int s = tid * 4;
    if (s + 3 < B) {
        const float4* xv = (const float4*)x;
        const float4 xa = xv[2 * tid];
        const float4 xb = xv[2 * tid + 1];
        float4 r;
        r.x = eval_sample(xa.x, xa.y, T0, T1, T2, T3);
        r.y = eval_sample(xa.z, xa.w, T0, T1, T2, T3);
        r.z = eval_sample(xb.x, xb.y, T0, T1, T2, T3);
        r.w = eval_sample(xb.z, xb.w, T0, T1, T2, T3);
        ((float4*)out)[tid] = r;
    } else if (s < B) {
        for (int i = s; i < B; ++i)
            out[i] = eval_sample(x[2 * i], x[2 * i + 1], T0, T1, T2, T3);
    }
}

} // anonymous namespace

extern "C" void kernel_launch(void* const* d_in, const int* in_sizes, int n_in,
                              void* d_out, int out_size, void* d_ws, size_t ws_size,
                              hipStream_t stream) {
    const float* x  = (const float*)d_in[0];   // [B,2] float32
    const float* qw = (const float*)d_in[1];   // [2,2] float32
    float* out = (float*)d_out;                // [B,1] float32
    float* T   = (float*)d_ws;                 // 16 floats scratch

    const int B = in_sizes[0] / 2;

    qnet_build_T<<<1, 32, 0, stream>>>(qw, T);

    if (B > 0 && (B % 4096) == 0) {
        qnet_main_tdm<<<B / 4096, 256, 0, stream>>>(x, T, out);
    } else {
        const int nthreads = (B + 3) / 4;
        const int blocks   = (nthreads + 255) / 256;
        qnet_main<<<blocks, 256, 0, stream>>>(x, T, out, B);
    }
}